// TemporalGNN_Regression_80023830659661
// MI455X (gfx1250) — compile-verified
//
#include <hip/hip_runtime.h>
#include <hip/hip_bf16.h>
#include <math.h>

// ---- problem constants (from reference) ----
#define NN 19     // nodes
#define KK 3      // kNN
#define HD 128    // hidden
#define BB 8      // batch
#define TT 256    // seq len
#define BN (BB * NN)   // 152 sequences
#define ROWS 160       // padded row count (10 tiles of 16)

typedef __attribute__((ext_vector_type(2))) float v2f;
typedef __attribute__((ext_vector_type(8))) float v8f;

// =====================================================================
// Kernel 1: cosine-similarity kNN graph (per batch). adj[b][i][j].
// =====================================================================
__global__ __launch_bounds__(256)
void build_graph_kernel(const float* __restrict__ x, float* __restrict__ adj) {
    __shared__ float xs[NN * TT];
    __shared__ float inv[NN];
    __shared__ float simL[NN * NN];
    __shared__ float adjL[NN * NN];
    const int b = blockIdx.x;

    for (int i = threadIdx.x; i < NN * TT; i += blockDim.x)
        xs[i] = x[(size_t)b * NN * TT + i];            // x[(b*19+n)*256 + t]
    __syncthreads();

    if (threadIdx.x < NN) {
        const float* r = xs + threadIdx.x * TT;
        float s = 0.f;
        for (int t = 0; t < TT; ++t) s += r[t] * r[t];
        inv[threadIdx.x] = 1.0f / (sqrtf(s) + 1e-12f);
    }
    for (int p = threadIdx.x; p < NN * NN; p += blockDim.x) adjL[p] = 0.f;
    __syncthreads();

    for (int p = threadIdx.x; p < NN * NN; p += blockDim.x) {
        int i = p / NN, j = p % NN;
        const float* ri = xs + i * TT;
        const float* rj = xs + j * TT;
        float s = 0.f;
        for (int t = 0; t < TT; ++t) s += ri[t] * rj[t];
        simL[p] = s * inv[i] * inv[j];
    }
    __syncthreads();

    // top-3 per row (strict > : lowest index wins ties, matches lax.top_k)
    if (threadIdx.x < NN) {
        const int i = threadIdx.x;
        bool used[NN];
        for (int j = 0; j < NN; ++j) used[j] = false;
        for (int k = 0; k < KK; ++k) {
            float best = -1e30f; int bj = 0;
            for (int j = 0; j < NN; ++j)
                if (!used[j] && simL[i * NN + j] > best) { best = simL[i * NN + j]; bj = j; }
            used[bj] = true;
            adjL[i * NN + bj] = best;
        }
    }
    __syncthreads();

    // symmetrize (max), self loops = 1
    for (int p = threadIdx.x; p < NN * NN; p += blockDim.x) {
        int i = p / NN, j = p % NN;
        float v = fmaxf(adjL[i * NN + j], adjL[j * NN + i]);
        if (i == j) v = 1.0f;
        adj[b * NN * NN + p] = v;
    }
}

// =====================================================================
// Kernel 2: GRU over T=256 steps. Single workgroup (sequential dep),
// h double-buffered in LDS (2 x 160 x 128 f32), matmul via f32 WMMA.
// =====================================================================
__global__ __launch_bounds__(512)
void gru_kernel(const float* __restrict__ x,
                const float* __restrict__ wi, const float* __restrict__ wh,
                const float* __restrict__ bi, const float* __restrict__ bh,
                float* __restrict__ Hout) {
    extern __shared__ float lds[];
    float* hb0 = lds;                 // ROWS*HD
    float* hb1 = lds + ROWS * HD;     // ROWS*HD

    for (int i = threadIdx.x; i < 2 * ROWS * HD; i += blockDim.x) lds[i] = 0.f;
    __syncthreads();

    const int lane = threadIdx.x & 31;
    const int wave = threadIdx.x >> 5;
    const int nw   = blockDim.x >> 5;   // 16 waves -> 80 jobs / 16 = 5 each
    const int kh   = lane >> 4;         // half (0/1)
    const int l15  = lane & 15;

    for (int t = 0; t < TT; ++t) {
        float* hc = (t & 1) ? hb1 : hb0;   // read
        float* hn = (t & 1) ? hb0 : hb1;   // write
        for (int job = wave; job < 80; job += nw) {
            const int mt = job % 10, nt = job / 10;
            const int m0 = mt * 16;
            const int col = nt * 16 + l15;          // 0..127 within a gate chunk
            v8f aR = {}, aZ = {}, aN = {};
            const float* arow = hc + (m0 + l15) * HD;
            for (int k = 0; k < HD; k += 4) {
                const int ka = k + 2 * kh;          // this lane's K pair
                v2f a; a.x = arow[ka]; a.y = arow[ka + 1];
                const float* w0 = wh + (size_t)ka * (3 * HD);
                const float* w1 = wh + (size_t)(ka + 1) * (3 * HD);
                v2f br, bz, bn;
                br.x = w0[col];             br.y = w1[col];
                bz.x = w0[col + HD];        bz.y = w1[col + HD];
                bn.x = w0[col + 2 * HD];    bn.y = w1[col + 2 * HD];
                aR = __builtin_amdgcn_wmma_f32_16x16x4_f32(false, a, false, br, (short)0, aR, false, false);
                aZ = __builtin_amdgcn_wmma_f32_16x16x4_f32(false, a, false, bz, (short)0, aZ, false, false);
                aN = __builtin_amdgcn_wmma_f32_16x16x4_f32(false, a, false, bn, (short)0, aN, false, false);
            }
            // fused gate math on the 16x16 D tile (rows m0..m0+15, cols of chunk)
#pragma unroll
            for (int i = 0; i < 8; ++i) {
                const int row = m0 + i + 8 * kh;
                const float xv = (row < BN) ? x[row * TT + t] : 0.0f;   // F=1
                const float gr = xv * wi[col]            + bi[col]            + aR[i] + bh[col];
                const float gz = xv * wi[col + HD]       + bi[col + HD]       + aZ[i] + bh[col + HD];
                const float gn = xv * wi[col + 2 * HD]   + bi[col + 2 * HD];
                const float hv = aN[i] + bh[col + 2 * HD];
                const float r = 1.0f / (1.0f + expf(-gr));
                const float z = 1.0f / (1.0f + expf(-gz));
                const float n = tanhf(gn + r * hv);
                const float hold = hc[row * HD + col];
                hn[row * HD + col] = (1.0f - z) * n + z * hold;
            }
        }
        __syncthreads();
    }
    // T=256 (even) -> final h lives in hb0
    for (int i = threadIdx.x; i < BN * HD; i += blockDim.x) Hout[i] = hb0[i];
}

// =====================================================================
// Kernel 3a: GINE aggregation. NOTE: all last-timestep graphs use adj[7]
// (reference's gidx = g % b quirk with g = bb*256 + 255).
// U[i] = h[i] + sum_j mask[j,i] * relu(h[j] + adj[j,i]*ew + eb)
// =====================================================================
__global__ __launch_bounds__(256)
void gine_agg_kernel(const float* __restrict__ Hin, const float* __restrict__ adj7,
                     const float* __restrict__ ew, const float* __restrict__ eb,
                     float* __restrict__ U) {
    __shared__ float hL[NN * HD];
    __shared__ float aL[NN * NN];
    const int b = blockIdx.x;
    for (int i = threadIdx.x; i < NN * HD; i += blockDim.x)
        hL[i] = Hin[b * NN * HD + i];
    for (int i = threadIdx.x; i < NN * NN; i += blockDim.x)
        aL[i] = adj7[i];
    __syncthreads();
    for (int p = threadIdx.x; p < NN * HD; p += blockDim.x) {
        const int i = p >> 7, c = p & 127;
        const float ewc = ew[c], ebc = eb[c];
        float s = hL[p];
        for (int j = 0; j < NN; ++j) {
            const float a = aL[i * NN + j];        // symmetric: adj[j][i]
            if (a != 0.0f) {
                const float m = hL[j * HD + c] + a * ewc + ebc;
                s += (m > 0.0f) ? m : 0.0f;
            }
        }
        U[b * NN * HD + p] = s;
    }
}

// =====================================================================
// Kernel 3b: Hout = relu(U @ W + bias), [160x128] @ [128x128], f32 WMMA.
// One wave per 16x16 tile; pad rows make all accesses unguarded.
// =====================================================================
__global__ __launch_bounds__(32)
void gine_mm_kernel(const float* __restrict__ U, const float* __restrict__ W,
                    const float* __restrict__ bias, float* __restrict__ Hout) {
    const int lane = threadIdx.x & 31;
    const int kh = lane >> 4, l15 = lane & 15;
    const int m0 = blockIdx.x * 16;
    const int n0 = blockIdx.y * 16;
    v8f acc = {};
    const float* arow = U + (m0 + l15) * HD;
    for (int k = 0; k < HD; k += 4) {
        const int ka = k + 2 * kh;
        v2f a; a.x = arow[ka]; a.y = arow[ka + 1];
        v2f bv;
        bv.x = W[ka * HD + n0 + l15];
        bv.y = W[(ka + 1) * HD + n0 + l15];
        acc = __builtin_amdgcn_wmma_f32_16x16x4_f32(false, a, false, bv, (short)0, acc, false, false);
    }
    const float bc = bias[n0 + l15];
#pragma unroll
    for (int i = 0; i < 8; ++i) {
        const int row = m0 + i + 8 * kh;
        const float v = acc[i] + bc;
        Hout[row * HD + n0 + l15] = (v > 0.0f) ? v : 0.0f;
    }
}

// =====================================================================
// Kernel 4: decoder, out[row] = h[row] . dec_w + dec_b
// =====================================================================
__global__ __launch_bounds__(160)
void decoder_kernel(const float* __restrict__ H2, const float* __restrict__ dw,
                    const float* __restrict__ db, float* __restrict__ out) {
    const int row = blockIdx.x * blockDim.x + threadIdx.x;
    if (row < BN) {
        float s = db[0];
        for (int c = 0; c < HD; ++c) s += H2[row * HD + c] * dw[c];
        out[row] = s;
    }
}

extern "C" void kernel_launch(void* const* d_in, const int* in_sizes, int n_in,
                              void* d_out, int out_size, void* d_ws, size_t ws_size,
                              hipStream_t stream) {
    const float* x       = (const float*)d_in[0];
    const float* gru_wi  = (const float*)d_in[1];
    const float* gru_wh  = (const float*)d_in[2];
    const float* gru_bi  = (const float*)d_in[3];
    const float* gru_bh  = (const float*)d_in[4];
    const float* gine_ew = (const float*)d_in[5];
    const float* gine_eb = (const float*)d_in[6];
    const float* gine_w  = (const float*)d_in[7];
    const float* gine_b  = (const float*)d_in[8];
    const float* dec_w   = (const float*)d_in[9];
    const float* dec_b   = (const float*)d_in[10];

    float* ws   = (float*)d_ws;
    float* adj  = ws;                          // 8*361
    float* Hbuf = ws + BB * NN * NN;           // 160*128 (padded)
    float* Ubuf = Hbuf + ROWS * HD;            // 160*128 (padded)

    // 1) kNN graphs
    build_graph_kernel<<<BB, 256, 0, stream>>>(x, adj);

    // 2) GRU (sequential over T; single WGP, h ping-pong in LDS)
    const size_t gru_lds = (size_t)2 * ROWS * HD * sizeof(float);  // 160 KB
    gru_kernel<<<1, 512, gru_lds, stream>>>(x, gru_wi, gru_wh, gru_bi, gru_bh, Hbuf);

    // 3) GINE x2 on the 8 last-timestep graphs (all use adj[7])
    for (int l = 0; l < 2; ++l) {
        gine_agg_kernel<<<BB, 256, 0, stream>>>(Hbuf, adj + 7 * NN * NN,
                                                gine_ew + l * HD, gine_eb + l * HD, Ubuf);
        gine_mm_kernel<<<dim3(10, 8), 32, 0, stream>>>(Ubuf, gine_w + l * HD * HD,
                                                       gine_b + l * HD, Hbuf);
    }

    // 4) decoder -> 152 floats
    decoder_kernel<<<1, 160, 0, stream>>>(Hbuf, dec_w, dec_b, (float*)d_out);
}